// AdaptiveMultiBoxLoss_75849122448025
// MI455X (gfx1250) — compile-verified
//
#include <hip/hip_runtime.h>
#include <hip/hip_bf16.h>
#include <math.h>

// Problem constants (from reference setup_inputs)
#define BATCH 64
#define NPRIOR 8732
#define NCLASS 81
#define BP (BATCH * NPRIOR)
#define PRIORS_PER_BLOCK 128                                   // 8 waves * 16 priors
#define NBLKX ((NPRIOR + PRIORS_PER_BLOCK - 1) / PRIORS_PER_BLOCK)  // 69
#define NBLK (NBLKX * BATCH)

typedef float v2f __attribute__((ext_vector_type(2)));
typedef float v8f __attribute__((ext_vector_type(8)));

__device__ __forceinline__ float smooth_l1(float d) {
  float a = fabsf(d);
  return (a < 1.0f) ? 0.5f * a * a : a - 0.5f;
}

__device__ __forceinline__ float wave_reduce_f(float x) {
#pragma unroll
  for (int o = 16; o > 0; o >>= 1) x += __shfl_xor(x, o, 32);
  return x;
}

__device__ __forceinline__ int wave_reduce_i(int x) {
#pragma unroll
  for (int o = 16; o > 0; o >>= 1) x += __shfl_xor(x, o, 32);
  return x;
}

// Wave-cooperative logsumexp over 81 classes for 16 priors.
// Lane l handles prior p0 + (l & 15); half = l>>4 selects classes {0,1} vs {2,3} mod 4.
// The exp-sum reduction runs on the matrix pipe: D = ones(16x4) x expvals(4x16) + C,
// so after 20 chained V_WMMA_F32_16X16X4_F32 ops, C[0] in lane l holds
// sum_{c=0..79} exp(conf[prior l%16][c] - max).  Class 80 is added scalar.
__device__ __forceinline__ float wave_lse16(const float* __restrict__ conf_b,
                                            int p_clamped, int lane) {
  const int half = lane >> 4;
  const float* rb = conf_b + (size_t)p_clamped * NCLASS;
  float v[40];
  float m = -3.402823466e+38f;
#pragma unroll
  for (int ch = 0; ch < 20; ++ch) {
    const int c = ch * 4 + 2 * half;
    const float a = rb[c];          // conf rows are only 4B aligned (81 floats) ->
    const float bv = rb[c + 1];     // keep b32 loads, L0 reuse keeps traffic minimal
    v[2 * ch] = a;
    v[2 * ch + 1] = bv;
    m = fmaxf(m, fmaxf(a, bv));
  }
  const float e80 = rb[80];
  m = fmaxf(m, e80);
  // lanes r and r+16 hold complementary halves of the same prior row
  m = fmaxf(m, __shfl_xor(m, 16, 32));

  v8f acc = {};
  v2f ones;
  ones.x = 1.0f;
  ones.y = 1.0f;
#pragma unroll
  for (int ch = 0; ch < 20; ++ch) {
    v2f bm;
    bm.x = __expf(v[2 * ch] - m);
    bm.y = __expf(v[2 * ch + 1] - m);
    // D[i][j] = sum_k ones[i][k] * bm[k][j] + C[i][j]  ->  per-prior column sums
    acc = __builtin_amdgcn_wmma_f32_16x16x4_f32(false, ones, false, bm,
                                                (short)0, acc, false, false);
  }
  const float s = acc[0] + __expf(e80 - m);  // C VGPR0, lane l -> N = l%16
  return m + __logf(s);
}

// Kernel 1: per-prior CE for teacher & student, positive-prior smooth-L1,
// per-block partial sums, per-batch positive counts.
__global__ void __launch_bounds__(256)
ce_kernel(const float* __restrict__ confT, const float* __restrict__ confS,
          const float* __restrict__ locT, const float* __restrict__ locS,
          const float* __restrict__ loc_t, const int* __restrict__ conf_t,
          float* __restrict__ lcT_out, float* __restrict__ lcS_out,
          float* __restrict__ partials, int* __restrict__ num_pos) {
  __shared__ float sh[8][4];
  __shared__ int shn[8];
  const int tid = threadIdx.x;
  const int wave = tid >> 5;
  const int lane = tid & 31;
  const int b = blockIdx.y;
  const int p0 = (blockIdx.x * 8 + wave) * 16;  // wave-uniform
  const int r = lane & 15;
  const int half = lane >> 4;

  float a_lT = 0.f, a_lS = 0.f, a_cT = 0.f, a_cS = 0.f;
  int a_np = 0;

  if (p0 < NPRIOR) {  // wave-uniform guard: EXEC stays all-ones around WMMA
    const int pr = p0 + r;
    const bool valid = pr < NPRIOR;
    const int pc = valid ? pr : (NPRIOR - 1);  // clamped, in-bounds loads
    const float* confT_b = confT + (size_t)b * NPRIOR * NCLASS;
    const float* confS_b = confS + (size_t)b * NPRIOR * NCLASS;
    const int gt = conf_t[b * NPRIOR + pc];
    const bool pos = gt > 0;

    {  // teacher CE
      const float lse = wave_lse16(confT_b, pc, lane);
      const float gtv = confT_b[(size_t)pc * NCLASS + gt];
      const float lc = lse - gtv;
      if (half == 0 && valid) {
        lcT_out[b * NPRIOR + pr] = lc;
        if (pos) {
          a_cT += lc;
          a_np += 1;
        }
      }
    }
    {  // student CE
      const float lse = wave_lse16(confS_b, pc, lane);
      const float gtv = confS_b[(size_t)pc * NCLASS + gt];
      const float lc = lse - gtv;
      if (half == 0 && valid) {
        lcS_out[b * NPRIOR + pr] = lc;
        if (pos) a_cS += lc;
      }
    }
    {  // smooth-L1 over positives: 16 priors * 4 coords = 64 floats, 2 per lane
      const int e = 2 * lane;             // even -> float2 stays 8B aligned
      const int pl = p0 + (lane >> 1);    // prior owning this coord pair
      const bool v2 = pl < NPRIOR;
      const int plc = v2 ? pl : (NPRIOR - 1);
      const int gtl = conf_t[b * NPRIOR + plc];
      const float w = (v2 && gtl > 0) ? 1.f : 0.f;
      const size_t off = ((size_t)b * NPRIOR + plc) * 4 + (e & 3);
      const float2 t = *(const float2*)(loc_t + off);
      const float2 xT = *(const float2*)(locT + off);
      const float2 xS = *(const float2*)(locS + off);
      a_lT += w * (smooth_l1(xT.x - t.x) + smooth_l1(xT.y - t.y));
      a_lS += w * (smooth_l1(xS.x - t.x) + smooth_l1(xS.y - t.y));
    }
  }

  a_lT = wave_reduce_f(a_lT);
  a_lS = wave_reduce_f(a_lS);
  a_cT = wave_reduce_f(a_cT);
  a_cS = wave_reduce_f(a_cS);
  a_np = wave_reduce_i(a_np);
  if (lane == 0) {
    sh[wave][0] = a_lT;
    sh[wave][1] = a_lS;
    sh[wave][2] = a_cT;
    sh[wave][3] = a_cS;
    shn[wave] = a_np;
  }
  __syncthreads();
  if (tid == 0) {
    float s0 = 0.f, s1 = 0.f, s2 = 0.f, s3 = 0.f;
    int sn = 0;
    for (int w = 0; w < 8; ++w) {
      s0 += sh[w][0];
      s1 += sh[w][1];
      s2 += sh[w][2];
      s3 += sh[w][3];
      sn += shn[w];
    }
    const int flat = b * gridDim.x + blockIdx.x;
    partials[flat * 4 + 0] = s0;
    partials[flat * 4 + 1] = s1;
    partials[flat * 4 + 2] = s2;
    partials[flat * 4 + 3] = s3;
    atomicAdd(&num_pos[b], sn);
  }
}

// Kernel 2: hard-negative mining. One block per (batch row, tensor).
// Exact: zero positives, bitonic-sort the row descending in LDS
// (8732 padded to 16384 floats = 64KB, fine on CDNA5's 320KB WGP LDS),
// then sum the top min(3*num_pos, P-1) entries.
__global__ void __launch_bounds__(1024)
mine_kernel(const float* __restrict__ lcT, const float* __restrict__ lcS,
            const int* __restrict__ conf_t, const int* __restrict__ num_pos,
            float* __restrict__ neg_acc) {
  __shared__ float s[16384];
  const int b = blockIdx.x;
  const int which = blockIdx.y;  // 0 = teacher, 1 = student
  const int tid = threadIdx.x;
  const float* lc = (which ? lcS : lcT) + (size_t)b * NPRIOR;
  const int* ct = conf_t + b * NPRIOR;

  for (int i = tid; i < 16384; i += 1024) {
    float val = -3.402823466e+38f;  // pad sorts to the tail (descending)
    if (i < NPRIOR) {
      val = lc[i];
      if (ct[i] > 0) val = 0.0f;  // positives excluded from mining
    }
    s[i] = val;
  }
  __syncthreads();

  for (int k = 2; k <= 16384; k <<= 1) {
    for (int j = k >> 1; j > 0; j >>= 1) {
      for (int i = tid; i < 16384; i += 1024) {
        const int ixj = i ^ j;
        if (ixj > i) {
          const float a = s[i];
          const float c = s[ixj];
          const bool up = (i & k) == 0;
          const bool sw = up ? (a < c) : (a > c);  // descending network
          if (sw) {
            s[i] = c;
            s[ixj] = a;
          }
        }
      }
      __syncthreads();
    }
  }

  const int np = num_pos[b];
  int kk = 3 * np;
  if (kk > NPRIOR - 1) kk = NPRIOR - 1;
  float part = 0.f;
  for (int i = tid; i < kk; i += 1024) part += s[i];
  part = wave_reduce_f(part);
  if ((tid & 31) == 0) atomicAdd(&neg_acc[which], part);
}

__global__ void zero_kernel(float* __restrict__ neg_acc, int* __restrict__ num_pos) {
  const int t = threadIdx.x;
  if (t < 8) neg_acc[t] = 0.f;
  if (t < BATCH) num_pos[t] = 0;
}

__global__ void __launch_bounds__(256)
finalize_kernel(const float* __restrict__ partials, const float* __restrict__ neg_acc,
                const int* __restrict__ num_pos, float* __restrict__ out) {
  __shared__ float sh[8][4];
  const int tid = threadIdx.x;
  float s0 = 0.f, s1 = 0.f, s2 = 0.f, s3 = 0.f;
  for (int i = tid; i < NBLK; i += 256) {
    s0 += partials[i * 4 + 0];
    s1 += partials[i * 4 + 1];
    s2 += partials[i * 4 + 2];
    s3 += partials[i * 4 + 3];
  }
  s0 = wave_reduce_f(s0);
  s1 = wave_reduce_f(s1);
  s2 = wave_reduce_f(s2);
  s3 = wave_reduce_f(s3);
  const int wave = tid >> 5, lane = tid & 31;
  if (lane == 0) {
    sh[wave][0] = s0;
    sh[wave][1] = s1;
    sh[wave][2] = s2;
    sh[wave][3] = s3;
  }
  __syncthreads();
  if (tid == 0) {
    float lT = 0.f, lS = 0.f, cT = 0.f, cS = 0.f;
    for (int w = 0; w < 8; ++w) {
      lT += sh[w][0];
      lS += sh[w][1];
      cT += sh[w][2];
      cS += sh[w][3];
    }
    int N = 0;
    for (int i = 0; i < BATCH; ++i) N += num_pos[i];
    const float Nf = (float)N;
    out[0] = lT / Nf;                 // loss_lT / N
    out[1] = (cT + neg_acc[0]) / Nf;  // loss_cT / N
    out[2] = lS / Nf;                 // loss_lS / N
    out[3] = (cS + neg_acc[1]) / Nf;  // loss_cS / N
  }
}

extern "C" void kernel_launch(void* const* d_in, const int* in_sizes, int n_in,
                              void* d_out, int out_size, void* d_ws, size_t ws_size,
                              hipStream_t stream) {
  // setup_inputs order: loc_dataT, conf_dataT, loc_dataS, conf_dataS, loc_t, conf_t
  const float* locT = (const float*)d_in[0];
  const float* confT = (const float*)d_in[1];
  const float* locS = (const float*)d_in[2];
  const float* confS = (const float*)d_in[3];
  const float* loc_t = (const float*)d_in[4];
  const int* conf_t = (const int*)d_in[5];
  float* out = (float*)d_out;

  // workspace layout (~4.5 MB): lcT, lcS, 8 accum floats, 64 int counts, partials
  float* lcT = (float*)d_ws;
  float* lcS = lcT + BP;
  float* neg_acc = lcS + BP;                 // [0]=cT_neg, [1]=cS_neg
  int* num_pos = (int*)(neg_acc + 8);        // [BATCH]
  float* partials = (float*)(num_pos + BATCH);  // [NBLK*4]

  hipLaunchKernelGGL(zero_kernel, dim3(1), dim3(128), 0, stream, neg_acc, num_pos);
  hipLaunchKernelGGL(ce_kernel, dim3(NBLKX, BATCH), dim3(256), 0, stream,
                     confT, confS, locT, locS, loc_t, conf_t,
                     lcT, lcS, partials, num_pos);
  hipLaunchKernelGGL(mine_kernel, dim3(BATCH, 2), dim3(1024), 0, stream,
                     lcT, lcS, conf_t, num_pos, neg_acc);
  hipLaunchKernelGGL(finalize_kernel, dim3(1), dim3(256), 0, stream,
                     partials, neg_acc, num_pos, out);
}